// DKVMN_4492535792110
// MI455X (gfx1250) — compile-verified
//
#include <hip/hip_runtime.h>
#include <hip/hip_bf16.h>
#include <math.h>

typedef __attribute__((ext_vector_type(2))) float v2f;
typedef __attribute__((ext_vector_type(8))) float v8f;

#define NUM_SKILLS 1000
#define DIM 64
#define SIZE_M 64
#define BATCH 64
#define TLEN 200
#define NROWS (BATCH * TLEN)   // 12800, divisible by 16

__device__ __forceinline__ float sigmf(float x) { return 1.0f / (1.0f + expf(-x)); }

// ---------------------------------------------------------------------------
// Phase A: per (b,t) row compute
//   logits = k @ Mk^T  -> w = softmax(logits)
//   epre   = v @ eW^T  -> e = sigmoid(epre + eb)
//   apre   = v @ aW^T  -> a = tanh(apre + ab)
// One block = 16 rows; 4 waves, each wave owns a 16-col tile of all 3 GEMMs.
// f32 WMMA 16x16x4, 16 k-steps over D=64. Exact f32 math.
// ---------------------------------------------------------------------------
__global__ __launch_bounds__(128) void dkvmn_phaseA(
    const int* __restrict__ skills, const int* __restrict__ resp,
    const float* __restrict__ k_emb, const float* __restrict__ v_emb,
    const float* __restrict__ Mk,
    const float* __restrict__ eW, const float* __restrict__ eb,
    const float* __restrict__ aW, const float* __restrict__ ab,
    float* __restrict__ wsW, float* __restrict__ wsE, float* __restrict__ wsA)
{
    __shared__ float lw[16][64];
    __shared__ float le[16][64];
    __shared__ float la[16][64];
    __shared__ float red[16][8];

    const int tid  = threadIdx.x;
    const int lane = tid & 31;
    const int wave = tid >> 5;
    const int hi   = lane >> 4;     // lane-half selects K offset (+2)
    const int lr   = lane & 15;     // A: row-in-tile, B: col-in-tile
    const int r0   = blockIdx.x * 16;

    // A rows (gathered embeddings) for this lane's matrix row
    const int r  = r0 + lr;
    const int sk = skills[r];
    int rp = resp[r];
    rp = (rp > -1) ? rp : 0;        // masked response (identity for {0,1})
    const int xi = sk + NUM_SKILLS * rp;
    const float* Ak = k_emb + (size_t)sk * DIM;
    const float* Av = v_emb + (size_t)xi * DIM;

    // B rows: weight row m0+lr, accessed as W^T (x @ W.T semantics)
    const int m0 = wave * 16;
    const float* Bm = Mk + (size_t)(m0 + lr) * DIM;
    const float* Be = eW + (size_t)(m0 + lr) * DIM;
    const float* Ba = aW + (size_t)(m0 + lr) * DIM;

    const int off0 = 2 * hi;

    v8f cw = {}; v8f ce = {}; v8f ca = {};
#pragma unroll
    for (int kk = 0; kk < 16; ++kk) {
        const int off = 4 * kk + off0;
        v2f akf = *(const v2f*)(Ak + off);
        v2f avf = *(const v2f*)(Av + off);
        v2f bmf = *(const v2f*)(Bm + off);
        v2f bef = *(const v2f*)(Be + off);
        v2f baf = *(const v2f*)(Ba + off);
        cw = __builtin_amdgcn_wmma_f32_16x16x4_f32(false, akf, false, bmf, (short)0, cw, false, false);
        ce = __builtin_amdgcn_wmma_f32_16x16x4_f32(false, avf, false, bef, (short)0, ce, false, false);
        ca = __builtin_amdgcn_wmma_f32_16x16x4_f32(false, avf, false, baf, (short)0, ca, false, false);
    }

    // C/D layout: lane l, vgpr i -> (M = i + 8*hi, N = lane%16)
#pragma unroll
    for (int i = 0; i < 8; ++i) {
        const int mrow = i + 8 * hi;
        lw[mrow][m0 + lr] = cw[i];
        le[mrow][m0 + lr] = ce[i];
        la[mrow][m0 + lr] = ca[i];
    }
    __syncthreads();

    // ---- softmax over 64 slots: 8 threads per row, LDS tree reductions ----
    const int sr = tid >> 3;        // row 0..15
    const int sc = tid & 7;         // 8-wide sub-column
    const int c0 = sc * 8;

    float mx = -1e30f;
#pragma unroll
    for (int j = 0; j < 8; ++j) mx = fmaxf(mx, lw[sr][c0 + j]);
    red[sr][sc] = mx;
    __syncthreads();
    float rmx = red[sr][0];
#pragma unroll
    for (int j = 1; j < 8; ++j) rmx = fmaxf(rmx, red[sr][j]);

    float ps = 0.0f;
    float ev[8];
#pragma unroll
    for (int j = 0; j < 8; ++j) {
        ev[j] = expf(lw[sr][c0 + j] - rmx);
        ps += ev[j];
    }
    __syncthreads();                // everyone done reading max partials
    red[sr][sc] = ps;
    __syncthreads();
    float rs = red[sr][0];
#pragma unroll
    for (int j = 1; j < 8; ++j) rs += red[sr][j];
    const float inv = 1.0f / rs;
    {
        const size_t rr = (size_t)(r0 + sr) * 64 + c0;
#pragma unroll
        for (int j = 0; j < 8; ++j) wsW[rr + j] = ev[j] * inv;
    }

    // ---- elementwise gates: all 128 threads ----
    for (int idx = tid; idx < 16 * 64; idx += 128) {
        const int rl = idx >> 6, c = idx & 63;
        const size_t rr = (size_t)(r0 + rl) * 64 + c;
        wsE[rr] = sigmf(le[rl][c] + eb[c]);
        wsA[rr] = tanhf(la[rl][c] + ab[c]);
    }
}

// ---------------------------------------------------------------------------
// Phase B: sequential DKVMN memory scan. One block per batch, 64 threads.
// Thread d owns column d of Mv in 64 REGISTERS (no LDS state traffic).
//   read_t[d] = sum_m w[m]*Mv[m][d]   (read BEFORE update)
//   Mv[m][d]  = Mv[m][d]*(1 - w[m]*e[d]) + w[m]*a[d]
// Next step's w/e/a are prefetched into registers to hide L2 latency.
// ---------------------------------------------------------------------------
__global__ __launch_bounds__(64) void dkvmn_phaseB(
    const float* __restrict__ Mv0,
    const float* __restrict__ wsW, const float* __restrict__ wsE,
    const float* __restrict__ wsA, float* __restrict__ wsR)
{
    __shared__ float wb[64];
    const int d = threadIdx.x;
    const int b = blockIdx.x;

    float mv[64];
#pragma unroll
    for (int m = 0; m < 64; ++m) mv[m] = Mv0[m * 64 + d];

    size_t row = (size_t)b * TLEN * 64;
    float wn = wsW[row + d];
    float en = wsE[row + d];
    float an = wsA[row + d];

    for (int t = 0; t < TLEN; ++t) {
        __syncthreads();            // previous step's wb reads complete
        wb[d] = wn;
        const float ed = en, ad = an;
        __syncthreads();            // wb visible to all
        if (t + 1 < TLEN) {         // uniform branch: prefetch next step
            wn = wsW[row + 64 + d];
            en = wsE[row + 64 + d];
            an = wsA[row + 64 + d];
        }
        float rd = 0.0f;
#pragma unroll
        for (int m = 0; m < 64; ++m) {
            const float wm = wb[m];         // LDS broadcast
            rd = fmaf(wm, mv[m], rd);
            const float we = wm * ed;
            mv[m] = fmaf(wm, ad, fmaf(-we, mv[m], mv[m]));
        }
        wsR[row + d] = rd;
        row += 64;
    }
}

// ---------------------------------------------------------------------------
// Phase C: f = tanh([read | k] @ fW^T + fb);  p = sigmoid(f . pW + pb)
// [12800 x 128] x [128 x 64] WMMA GEMM, 32 k-steps (16 from reads, 16 from k).
// ---------------------------------------------------------------------------
__global__ __launch_bounds__(128) void dkvmn_phaseC(
    const int* __restrict__ skills,
    const float* __restrict__ k_emb,
    const float* __restrict__ fW, const float* __restrict__ fb,
    const float* __restrict__ pW, const float* __restrict__ pb,
    const float* __restrict__ wsR,
    float* __restrict__ out)
{
    __shared__ float fs[16][64];
    __shared__ float red[16][8];
    const int tid  = threadIdx.x;
    const int lane = tid & 31;
    const int wave = tid >> 5;
    const int hi   = lane >> 4;
    const int lr   = lane & 15;
    const int r0   = blockIdx.x * 16;

    const int r  = r0 + lr;
    const int sk = skills[r];
    const float* Ar  = wsR + (size_t)r * 64;          // concat cols 0..63
    const float* Akk = k_emb + (size_t)sk * DIM;      // concat cols 64..127
    const int m0 = wave * 16;
    const float* Bf = fW + (size_t)(m0 + lr) * 128;   // fW row = output dim
    const int off0 = 2 * hi;

    v8f cf = {};
#pragma unroll
    for (int kk = 0; kk < 16; ++kk) {
        const int off = 4 * kk + off0;
        v2f a = *(const v2f*)(Ar + off);
        v2f w = *(const v2f*)(Bf + off);
        cf = __builtin_amdgcn_wmma_f32_16x16x4_f32(false, a, false, w, (short)0, cf, false, false);
    }
#pragma unroll
    for (int kk = 0; kk < 16; ++kk) {
        const int off = 4 * kk + off0;
        v2f a = *(const v2f*)(Akk + off);
        v2f w = *(const v2f*)(Bf + 64 + off);
        cf = __builtin_amdgcn_wmma_f32_16x16x4_f32(false, a, false, w, (short)0, cf, false, false);
    }

#pragma unroll
    for (int i = 0; i < 8; ++i) {
        const int mrow = i + 8 * hi;
        fs[mrow][m0 + lr] = tanhf(cf[i] + fb[m0 + lr]);
    }
    __syncthreads();

    // Prediction head: 8 threads per row, LDS reduction
    const int sr = tid >> 3;
    const int sc = tid & 7;
    const int c0 = sc * 8;
    float s = 0.0f;
#pragma unroll
    for (int j = 0; j < 8; ++j) s = fmaf(fs[sr][c0 + j], pW[c0 + j], s);
    red[sr][sc] = s;
    __syncthreads();
    if (sc == 0) {
        float tot = pb[0];
#pragma unroll
        for (int j = 0; j < 8; ++j) tot += red[sr][j];
        const float p = sigmf(tot);
        const int rr = r0 + sr;
        const int bb = rr / TLEN, tt = rr % TLEN;
        if (tt > 0) out[(size_t)bb * (TLEN - 1) + (tt - 1)] = p;
    }
}

extern "C" void kernel_launch(void* const* d_in, const int* in_sizes, int n_in,
                              void* d_out, int out_size, void* d_ws, size_t ws_size,
                              hipStream_t stream) {
    const int*   skills = (const int*)d_in[0];
    const int*   resp   = (const int*)d_in[1];
    const float* k_emb  = (const float*)d_in[2];
    const float* v_emb  = (const float*)d_in[3];
    const float* Mk     = (const float*)d_in[4];
    const float* Mv0    = (const float*)d_in[5];
    const float* fW     = (const float*)d_in[6];
    const float* fb     = (const float*)d_in[7];
    const float* eW     = (const float*)d_in[8];
    const float* eb     = (const float*)d_in[9];
    const float* aW     = (const float*)d_in[10];
    const float* ab     = (const float*)d_in[11];
    const float* pW     = (const float*)d_in[12];
    const float* pb     = (const float*)d_in[13];
    float* out = (float*)d_out;

    float* ws = (float*)d_ws;
    const size_t N = (size_t)NROWS * 64;   // 819200 floats per array
    float* wsW = ws;
    float* wsE = ws + N;
    float* wsA = ws + 2 * N;
    float* wsR = ws + 3 * N;

    dkvmn_phaseA<<<NROWS / 16, 128, 0, stream>>>(skills, resp, k_emb, v_emb, Mk,
                                                 eW, eb, aW, ab, wsW, wsE, wsA);
    dkvmn_phaseB<<<BATCH, 64, 0, stream>>>(Mv0, wsW, wsE, wsA, wsR);
    dkvmn_phaseC<<<NROWS / 16, 128, 0, stream>>>(skills, k_emb, fW, fb, pW, pb, wsR, out);
}